// AtlasFreeBrainTransformer_48369921688169
// MI455X (gfx1250) — compile-verified
//
#include <hip/hip_runtime.h>
#include <math.h>

// ---------------------------------------------------------------------------
// CDNA5 (gfx1250) fp32 WMMA transformer pipeline.
// All GEMMs + attention use V_WMMA_F32_16X16X4_F32 (wave32, 16x16 tile/wave).
// ---------------------------------------------------------------------------

typedef float v2f __attribute__((ext_vector_type(2)));
typedef float v8f __attribute__((ext_vector_type(8)));

#define NMAXPAD 4064   // 254 * 16, >= max token count 4050
#define NPOS    4050   // 15*18*15 pooled positions

__device__ static inline v8f wmma_f32(v2f a, v2f b, v8f c) {
    // D = A(16x4) * B(4x16) + C(16x16), fp32
    return __builtin_amdgcn_wmma_f32_16x16x4_f32(
        false, a, false, b, (short)0, c, false, false);
}

// ---------------------------------------------------------------------------
// Mask + ordered compaction of valid pooled windows. Single block.
// mask[i] = OR over both batches and 27 voxels of (C > 0).
// ---------------------------------------------------------------------------
__global__ void mask_scan_kernel(const int* __restrict__ C,
                                 int* __restrict__ pos,
                                 int* __restrict__ meta) {
    __shared__ unsigned char sm[NPOS];
    int tid = threadIdx.x;
    for (int i = tid; i < NPOS; i += blockDim.x) {
        int pz = i % 15, py = (i / 15) % 18, px = i / 270;
        int any = 0;
        for (int b = 0; b < 2; ++b)
            for (int dx = 0; dx < 3; ++dx)
                for (int dy = 0; dy < 3; ++dy)
                    for (int dz = 0; dz < 3; ++dz) {
                        int x = 3 * px + dx, y = 3 * py + dy, z = 3 * pz + dz;
                        int c = C[((b * 45 + x) * 54 + y) * 45 + z];
                        any |= (c > 0);
                    }
        sm[i] = (unsigned char)any;
    }
    __syncthreads();
    if (tid == 0) {
        int cnt = 0;
        for (int i = 0; i < NPOS; ++i)
            if (sm[i]) pos[cnt++] = i;
        meta[0] = cnt;                  // N valid tokens
        meta[1] = (cnt + 15) & ~15;     // padded to 16
    }
}

// ---------------------------------------------------------------------------
// Generic wave-per-16x16-tile GEMM:  D = act(A @ B (+bias)) (+= old D)
// A: row-major (M x K), lda (even), per-batch stride sA.
// BTRANS=true : Bw row-major (N x K) used as B^T (ldb even) -> v2f loads.
// BTRANS=false: Bw row-major (K x N), column-strided scalar loads.
// M either fixed or read from meta[0] (runtime token count).
// K must be a multiple of 4. N (grid.x*16) must exactly tile the weights.
// Fragment layouts per CDNA5 ISA 7.12.2 (wave32):
//   A: lane l(0..15)=row, half=lane>>4 selects K pair;  B mirrored (col=l);
//   C/D: c[r] -> row = mb + r + 8*half, col = nb + l.
// ---------------------------------------------------------------------------
template <bool BTRANS>
__global__ void gemm16(const float* __restrict__ A, int lda, long sA,
                       const float* __restrict__ Bw, int ldb,
                       const float* __restrict__ bias,
                       float* __restrict__ D, int ldd, long sD,
                       const int* __restrict__ Mptr, int Mfixed,
                       int K, int accum, int act) {
    int M = Mptr ? Mptr[0] : Mfixed;
    int mb = blockIdx.y << 4;
    if (mb >= M) return;
    int nb = blockIdx.x << 4;
    const float* Ab = A + (long)blockIdx.z * sA;
    float*       Db = D + (long)blockIdx.z * sD;

    int lane = threadIdx.x;
    int half = lane >> 4;
    int l    = lane & 15;

    // A fragments: 8-byte vector loads (lda even, k pair-aligned)
    const v2f* ap = (const v2f*)(Ab + (long)(mb + l) * lda);

    v8f c = {};
    if (BTRANS) {
        const v2f* bp = (const v2f*)(Bw + (long)(nb + l) * ldb);
        for (int k = 0; k < K; k += 4) {
            int kh = (k >> 1) + half;
            c = wmma_f32(ap[kh], bp[kh], c);
        }
    } else {
        const float* bp = Bw + nb + l;
        for (int k = 0; k < K; k += 4) {
            int k0 = k + 2 * half;
            v2f b = { bp[(long)k0 * ldb], bp[(long)(k0 + 1) * ldb] };
            c = wmma_f32(ap[(k >> 1) + half], b, c);
        }
    }

    float bs = bias ? bias[nb + l] : 0.0f;
    for (int r = 0; r < 8; ++r) {
        int m = mb + r + 8 * half;
        if (m < M) {
            float v = c[r] + bs;
            if (act == 1) v = fmaxf(v, 0.0f);                                   // relu
            else if (act == 2) v = 0.5f * v * (1.0f + erff(v * 0.70710678f));   // exact gelu
            long idx = (long)m * ldd + nb + l;
            if (accum) v += Db[idx];
            Db[idx] = v;
        }
    }
}

// ---------------------------------------------------------------------------
// Gather embedded ROI rows through C and 3x3x3 sum-pool into token j.
// F_pad[0] is implicit zeros: index c contributes F_emb[c-1] iff c > 0.
// ---------------------------------------------------------------------------
__global__ void gather_pool(const int* __restrict__ C,
                            const float* __restrict__ Femb,
                            const int* __restrict__ pos,
                            const int* __restrict__ meta,
                            float* __restrict__ tok) {
    int N = meta[0];
    int j = blockIdx.x;
    if (j >= N) return;
    int b = blockIdx.y, d = threadIdx.x;
    __shared__ int ci[27];
    if (d < 27) {
        int p = pos[j];
        int pz = p % 15, py = (p / 15) % 18, px = p / 270;
        int dz = d % 3, dy = (d / 3) % 3, dx = d / 9;
        int xx = 3 * px + dx, yy = 3 * py + dy, zz = 3 * pz + dz;
        ci[d] = C[((b * 45 + xx) * 54 + yy) * 45 + zz];
    }
    __syncthreads();
    float sum = 0.0f;
    for (int i = 0; i < 27; ++i) {
        int c = ci[i];
        if (c > 0) sum += Femb[((long)b * 400 + (c - 1)) * 128 + d];
    }
    tok[((long)b * NMAXPAD + j) * 128 + d] = sum;
}

// ---------------------------------------------------------------------------
// Flash attention per (q-tile 16, head, batch), one wave per block.
// S-tile via 8x wmma (K=32); online softmax with shfl row reductions
// (S lane layout: row = r + 8*half, col/key = kb + l);
// P relayout C->A via 16x17 LDS tile; P@V via 8x wmma per key tile.
// Invalid key columns (>= N) forced to -1e30 -> exp == 0.
// ---------------------------------------------------------------------------
__global__ void attn_flash(const float* __restrict__ qkv,
                           float* __restrict__ o,
                           const int* __restrict__ meta) {
    int N = meta[0];
    int qb = blockIdx.x << 4;
    if (qb >= N) return;
    int ntiles = (N + 15) >> 4;
    int h = blockIdx.y, b = blockIdx.z;
    const float* base = qkv + (long)b * NMAXPAD * 384;

    int lane = threadIdx.x;
    int half = lane >> 4;
    int l    = lane & 15;
    const float scale = 0.17677669529663688f;   // 1/sqrt(32)

    v2f qa[8];
    {
        const v2f* qp = (const v2f*)(base + (long)(qb + l) * 384 + h * 32);
        for (int s = 0; s < 8; ++s) {
            v2f q = qp[2 * s + half];
            qa[s].x = q.x * scale;
            qa[s].y = q.y * scale;
        }
    }

    float mi[8], li[8];
    v8f acc0 = {}, acc1 = {};
    for (int r = 0; r < 8; ++r) { mi[r] = -1e30f; li[r] = 0.0f; }

    __shared__ float P[16][17];

    for (int kt = 0; kt < ntiles; ++kt) {
        int kb = kt << 4;
        // S = (Q*scale) @ K^T
        v8f sa = {};
        const v2f* kp = (const v2f*)(base + (long)(kb + l) * 384 + 128 + h * 32);
        for (int s = 0; s < 8; ++s)
            sa = wmma_f32(qa[s], kp[2 * s + half], sa);

        bool kval = (kb + l) < N;
        for (int r = 0; r < 8; ++r) {
            float sv = kval ? sa[r] : -1e30f;
            float mx = sv;
            mx = fmaxf(mx, __shfl_xor(mx, 1, 32));
            mx = fmaxf(mx, __shfl_xor(mx, 2, 32));
            mx = fmaxf(mx, __shfl_xor(mx, 4, 32));
            mx = fmaxf(mx, __shfl_xor(mx, 8, 32));
            float mnew  = fmaxf(mi[r], mx);
            float p     = __expf(sv - mnew);
            float alpha = __expf(mi[r] - mnew);
            float rs = p;
            rs += __shfl_xor(rs, 1, 32);
            rs += __shfl_xor(rs, 2, 32);
            rs += __shfl_xor(rs, 4, 32);
            rs += __shfl_xor(rs, 8, 32);
            li[r] = li[r] * alpha + rs;
            mi[r] = mnew;
            acc0[r] *= alpha;
            acc1[r] *= alpha;
            P[r + 8 * half][l] = p;
        }
        __syncthreads();
        // O += P @ V   (two 16-col output tiles for dh = 32)
        for (int s2 = 0; s2 < 4; ++s2) {
            int k0 = 4 * s2 + 2 * half;
            v2f pa = { P[l][k0], P[l][k0 + 1] };
            const float* vp = base + (long)(kb + k0) * 384 + 256 + h * 32 + l;
            v2f b0 = { vp[0],  vp[384] };
            v2f b1 = { vp[16], vp[16 + 384] };
            acc0 = wmma_f32(pa, b0, acc0);
            acc1 = wmma_f32(pa, b1, acc1);
        }
        __syncthreads();
    }

    for (int r = 0; r < 8; ++r) {
        int m = qb + r + 8 * half;
        if (m < N) {
            float inv = 1.0f / li[r];
            float* op = o + ((long)b * NMAXPAD + m) * 128 + h * 32 + l;
            op[0]  = acc0[r] * inv;
            op[16] = acc1[r] * inv;
        }
    }
}

// ---------------------------------------------------------------------------
// x = LayerNorm(x + f) * g + b, one block (128 threads) per token row.
// ---------------------------------------------------------------------------
__global__ void residual_ln(float* __restrict__ x, const float* __restrict__ f,
                            const float* __restrict__ g, const float* __restrict__ bt,
                            const int* __restrict__ meta) {
    int N = meta[0];
    int row = blockIdx.x;
    if (row >= N) return;
    int b = blockIdx.y, t = threadIdx.x;
    long off = ((long)b * NMAXPAD + row) * 128 + t;
    float v = x[off] + f[off];

    __shared__ float red[4];
    int wid = t >> 5;
    float s = v;
    for (int o = 16; o > 0; o >>= 1) s += __shfl_xor(s, o, 32);
    if ((t & 31) == 0) red[wid] = s;
    __syncthreads();
    float mean = (red[0] + red[1] + red[2] + red[3]) * (1.0f / 128.0f);
    __syncthreads();
    float d = v - mean;
    float s2 = d * d;
    for (int o = 16; o > 0; o >>= 1) s2 += __shfl_xor(s2, o, 32);
    if ((t & 31) == 0) red[wid] = s2;
    __syncthreads();
    float var = (red[0] + red[1] + red[2] + red[3]) * (1.0f / 128.0f);
    x[off] = d * rsqrtf(var + 1e-5f) * g[t] + bt[t];
}

// ---------------------------------------------------------------------------
// hm[b][d] = mean over valid tokens of x
// ---------------------------------------------------------------------------
__global__ void mean_pool(const float* __restrict__ x,
                          const int* __restrict__ meta,
                          float* __restrict__ hm) {
    int N = meta[0];
    int b = blockIdx.x, d = threadIdx.x;
    const float* xb = x + (long)b * NMAXPAD * 128 + d;
    float s = 0.0f;
    for (int i = 0; i < N; ++i) s += xb[(long)i * 128];
    hm[b * 128 + d] = s / (float)N;
}

// ---------------------------------------------------------------------------
// out = relu(hm @ c_w1 + b1) @ c_w2 + b2   (tiny: scalar per batch)
// ---------------------------------------------------------------------------
__global__ void classifier(const float* __restrict__ hm,
                           const float* __restrict__ w1, const float* __restrict__ b1,
                           const float* __restrict__ w2, const float* __restrict__ b2,
                           float* __restrict__ out) {
    int b = blockIdx.x, t = threadIdx.x;
    __shared__ float c1[64];
    float acc = b1[t];
    for (int k = 0; k < 128; ++k) acc += hm[b * 128 + k] * w1[k * 64 + t];
    c1[t] = fmaxf(acc, 0.0f);
    __syncthreads();
    if (t < 2) {
        float a = b2[t];
        for (int k = 0; k < 64; ++k) a += c1[k] * w2[k * 2 + t];
        out[b * 2 + t] = a;
    }
}

// ---------------------------------------------------------------------------
extern "C" void kernel_launch(void* const* d_in, const int* in_sizes, int n_in,
                              void* d_out, int out_size, void* d_ws, size_t ws_size,
                              hipStream_t stream) {
    (void)in_sizes; (void)n_in; (void)out_size; (void)ws_size;

    const float* F_roi = (const float*)d_in[0];
    const int*   C     = (const int*)  d_in[1];
    const float* re_w1 = (const float*)d_in[2];
    const float* re_b1 = (const float*)d_in[3];
    const float* re_w2 = (const float*)d_in[4];
    const float* re_b2 = (const float*)d_in[5];
    const float* c_w1  = (const float*)d_in[30];
    const float* c_b1  = (const float*)d_in[31];
    const float* c_w2  = (const float*)d_in[32];
    const float* c_b2  = (const float*)d_in[33];

    char* ws = (char*)d_ws;
    int*   meta = (int*)  (ws + 0);
    int*   pos  = (int*)  (ws + 256);
    float* H1   = (float*)(ws + 16640);      // 800 x 256
    float* Femb = (float*)(ws + 835840);     // 800 x 128
    float* X    = (float*)(ws + 1245440);    // 2 x 4064 x 128
    float* QKV  = (float*)(ws + 5406976);    // 2 x 4064 x 384 (reused for attn proj)
    float* O    = (float*)(ws + 17891584);   // 2 x 4064 x 128 (attn out / FF out)
    float* MID  = (float*)(ws + 22053120);   // 2 x 4064 x 256 (FF chunk)
    float* HM   = (float*)(ws + 30376192);   // 2 x 128
    float* OUT  = (float*)d_out;

    const long sX = (long)NMAXPAD * 128;
    const long sQ = (long)NMAXPAD * 384;
    const long sM = (long)NMAXPAD * 256;

    // 1) valid mask + compaction
    mask_scan_kernel<<<1, 1024, 0, stream>>>(C, pos, meta);

    // 2) ROI embedding MLP: (800x512)@(512x256) relu, (800x256)@(256x128)
    gemm16<false><<<dim3(16, 50, 1), 32, 0, stream>>>(F_roi, 512, 0, re_w1, 256, re_b1,
                                                      H1, 256, 0, nullptr, 800, 512, 0, 1);
    gemm16<false><<<dim3(8, 50, 1), 32, 0, stream>>>(H1, 256, 0, re_w2, 128, re_b2,
                                                     Femb, 128, 0, nullptr, 800, 256, 0, 0);

    // 3) gather + 3x3x3 sum-pool into compacted tokens
    gather_pool<<<dim3(4050, 2), 128, 0, stream>>>(C, Femb, pos, meta, X);

    // 4) two encoder layers
    for (int L = 0; L < 2; ++L) {
        int base = 6 + 12 * L;
        const float* in_w  = (const float*)d_in[base + 0];
        const float* in_b  = (const float*)d_in[base + 1];
        const float* out_w = (const float*)d_in[base + 2];
        const float* out_b = (const float*)d_in[base + 3];
        const float* ln1_g = (const float*)d_in[base + 4];
        const float* ln1_b = (const float*)d_in[base + 5];
        const float* ff1_w = (const float*)d_in[base + 6];
        const float* ff1_b = (const float*)d_in[base + 7];
        const float* ff2_w = (const float*)d_in[base + 8];
        const float* ff2_b = (const float*)d_in[base + 9];
        const float* ln2_g = (const float*)d_in[base + 10];
        const float* ln2_b = (const float*)d_in[base + 11];

        // QKV = X @ in_w^T + in_b
        gemm16<true><<<dim3(24, 254, 2), 32, 0, stream>>>(X, 128, sX, in_w, 128, in_b,
                                                          QKV, 384, sQ, meta, 0, 128, 0, 0);
        // flash attention -> O
        attn_flash<<<dim3(254, 4, 2), 32, 0, stream>>>(QKV, O, meta);
        // proj = O @ out_w^T + out_b  (written into QKV buffer, now free)
        gemm16<true><<<dim3(8, 254, 2), 32, 0, stream>>>(O, 128, sX, out_w, 128, out_b,
                                                         QKV, 128, sX, meta, 0, 128, 0, 0);
        // X = LN(X + proj)
        residual_ln<<<dim3(4050, 2), 128, 0, stream>>>(X, QKV, ln1_g, ln1_b, meta);

        // FF, chunked over the 2048 hidden dim (8 chunks of 256)
        for (int c = 0; c < 8; ++c) {
            gemm16<true><<<dim3(16, 254, 2), 32, 0, stream>>>(
                X, 128, sX, ff1_w + (long)c * 256 * 128, 128, ff1_b + c * 256,
                MID, 256, sM, meta, 0, 128, 0, 2 /*gelu*/);
            gemm16<true><<<dim3(8, 254, 2), 32, 0, stream>>>(
                MID, 256, sM, ff2_w + c * 256, 2048, (c == 0) ? ff2_b : nullptr,
                O, 128, sX, meta, 0, 256, (c > 0) ? 1 : 0, 0);
        }
        // X = LN(X + FF)
        residual_ln<<<dim3(4050, 2), 128, 0, stream>>>(X, O, ln2_g, ln2_b, meta);
    }

    // 5) mean pool + classifier
    mean_pool<<<2, 128, 0, stream>>>(X, meta, HM);
    classifier<<<2, 64, 0, stream>>>(HM, c_w1, c_b1, c_w2, c_b2, OUT);
}